// GraphConv_12721693131118
// MI455X (gfx1250) — compile-verified
//
#include <hip/hip_runtime.h>

#define DIM   64
#define SLOTS 4                  // N_HOPS + 1 embedding slots per node
#define CHUNK 128                // edges staged per wave per buffer (3x b128 async = 512B each)
#define WAVES_PER_BLOCK 8
#define BLOCK 256

// ---------------------------------------------------------------------------
// Stage CHUNK edge triples (vals/rows/cols) into this wave's LDS buffer using
// the CDNA5 async global->LDS path (tracked by ASYNCcnt, zero VGPR footprint).
// Each lane moves 16B per array: 32 lanes x 16B = 512B = 128 f32/i32 = CHUNK.
// s_wait_dscnt 0x0 first: guarantee our previous ds_load reads of this buffer
// have drained before the async writes can land (WAR on LDS).
// ---------------------------------------------------------------------------
__device__ __forceinline__ void async_stage(const float* vals, const int* rows,
                                            const int* cols, unsigned voff,
                                            unsigned ldsV, unsigned ldsR, unsigned ldsC)
{
    asm volatile("s_wait_dscnt 0x0" ::: "memory");
    asm volatile("global_load_async_to_lds_b128 %0, %1, %2"
                 :: "v"(ldsV), "v"(voff), "s"(vals) : "memory");
    asm volatile("global_load_async_to_lds_b128 %0, %1, %2"
                 :: "v"(ldsR), "v"(voff), "s"(rows) : "memory");
    asm volatile("global_load_async_to_lds_b128 %0, %1, %2"
                 :: "v"(ldsC), "v"(voff), "s"(cols) : "memory");
}

// Native f32 scatter-add (no-return form, STOREcnt). Avoids any CAS fallback.
__device__ __forceinline__ void atomic_fadd(float* p, float v)
{
    asm volatile("global_atomic_add_f32 %0, %1, off" :: "v"(p), "v"(v) : "memory");
}

// ---------------------------------------------------------------------------
// out layout: [node][slot][64] f32, node stride = SLOTS*DIM = 256 floats.
// Hop h: dst slot h accumulates v * out[col][h-1][:]. One wave per edge:
// lane L owns dims {2L, 2L+1} -> 256B coalesced gather + coalesced atomics.
// ---------------------------------------------------------------------------
__global__ void __launch_bounds__(BLOCK)
spmm_hop(const float* __restrict__ vals, const int* __restrict__ rows,
         const int* __restrict__ cols, float* __restrict__ out,
         int nnz, int src_slot, int dst_slot)
{
    __shared__ float sval[WAVES_PER_BLOCK][2][CHUNK];
    __shared__ int   srow[WAVES_PER_BLOCK][2][CHUNK];
    __shared__ int   scol[WAVES_PER_BLOCK][2][CHUNK];

    const int lane    = threadIdx.x & 31;
    const int w       = threadIdx.x >> 5;
    const int wave_id = blockIdx.x * WAVES_PER_BLOCK + w;
    const int nwaves  = gridDim.x * WAVES_PER_BLOCK;
    const int nchunks = nnz / CHUNK;

    const float* src = out + src_slot * DIM;
    float*       dst = out + dst_slot * DIM;

    int chunk = wave_id;
    int buf   = 0;
    if (chunk < nchunks) {
        unsigned voff = (unsigned)chunk * (CHUNK * 4u) + (unsigned)lane * 16u;
        async_stage(vals, rows, cols, voff,
                    (unsigned)(uintptr_t)&sval[w][0][lane * 4],
                    (unsigned)(uintptr_t)&srow[w][0][lane * 4],
                    (unsigned)(uintptr_t)&scol[w][0][lane * 4]);
    }

    while (chunk < nchunks) {
        const int next = chunk + nwaves;
        if (next < nchunks) {
            // Prefetch next chunk into the other buffer, then wait only for
            // the 3 oldest async ops (in-order completion) -> current buffer ready.
            const int nb = buf ^ 1;
            unsigned voff = (unsigned)next * (CHUNK * 4u) + (unsigned)lane * 16u;
            async_stage(vals, rows, cols, voff,
                        (unsigned)(uintptr_t)&sval[w][nb][lane * 4],
                        (unsigned)(uintptr_t)&srow[w][nb][lane * 4],
                        (unsigned)(uintptr_t)&scol[w][nb][lane * 4]);
            asm volatile("s_wait_asynccnt 0x3" ::: "memory");
        } else {
            asm volatile("s_wait_asynccnt 0x0" ::: "memory");
        }

        #pragma unroll 4
        for (int e = 0; e < CHUNK; ++e) {
            const float v = sval[w][buf][e];   // wave-uniform LDS broadcasts
            const int   r = srow[w][buf][e];
            const int   c = scol[w][buf][e];

            if ((e & 7) == 0 && e + 8 < CHUNK) {        // pre-touch future gather row
                const int pc = scol[w][buf][e + 8];
                __builtin_prefetch(src + (size_t)pc * (SLOTS * DIM), 0, 0);
            }

            const float2 g = *(const float2*)(src + (size_t)c * (SLOTS * DIM) + lane * 2);
            float* p = dst + (size_t)r * (SLOTS * DIM) + lane * 2;
            atomic_fadd(p,     v * g.x);
            atomic_fadd(p + 1, v * g.y);
        }
        chunk = next;
        buf ^= 1;
    }

    // Tail edges (nnz % CHUNK), plain loads, still wave-cooperative.
    for (int e = nchunks * CHUNK + wave_id; e < nnz; e += nwaves) {
        const float v = vals[e];
        const int   r = rows[e];
        const int   c = cols[e];
        const float2 g = *(const float2*)(src + (size_t)c * (SLOTS * DIM) + lane * 2);
        float* p = dst + (size_t)r * (SLOTS * DIM) + lane * 2;
        atomic_fadd(p,     v * g.x);
        atomic_fadd(p + 1, v * g.y);
    }
}

// ---------------------------------------------------------------------------
// Slot 0 = concat(user_embed, item_embed); slots 1..3 = 0 (atomic targets).
// Fully b128 vectorized; rewrites everything each call => replay-deterministic.
// ---------------------------------------------------------------------------
__global__ void __launch_bounds__(BLOCK)
init_out(const float4* __restrict__ user4, const float4* __restrict__ item4,
         float4* __restrict__ out4, int n_users, int n_nodes)
{
    const long total = (long)n_nodes * (SLOTS * DIM / 4);   // 64 float4 per node
    for (long i = blockIdx.x * (long)blockDim.x + threadIdx.x; i < total;
         i += (long)gridDim.x * blockDim.x) {
        const long node = i >> 6;          // 64 float4 per node
        const int  q    = (int)(i & 63);   // float4 index within node
        float4 val = make_float4(0.f, 0.f, 0.f, 0.f);
        if (q < DIM / 4) {                 // slot 0: copy embedding
            val = (node < n_users) ? user4[node * (DIM / 4) + q]
                                   : item4[(node - n_users) * (DIM / 4) + q];
        }
        out4[i] = val;
    }
}

extern "C" void kernel_launch(void* const* d_in, const int* in_sizes, int n_in,
                              void* d_out, int out_size, void* d_ws, size_t ws_size,
                              hipStream_t stream)
{
    const float* user = (const float*)d_in[0];
    const float* item = (const float*)d_in[1];
    const float* vals = (const float*)d_in[2];
    const int*   rows = (const int*)d_in[3];
    const int*   cols = (const int*)d_in[4];

    const int n_users = in_sizes[0] / DIM;
    const int n_items = in_sizes[1] / DIM;
    const int n_nodes = n_users + n_items;
    const int nnz     = in_sizes[2];

    float* out = (float*)d_out;

    init_out<<<2048, BLOCK, 0, stream>>>((const float4*)user, (const float4*)item,
                                         (float4*)out, n_users, n_nodes);
    for (int h = 1; h < SLOTS; ++h) {
        spmm_hop<<<1280, BLOCK, 0, stream>>>(vals, rows, cols, out, nnz, h - 1, h);
    }
}